// SSTInputLayer_70059506532501
// MI455X (gfx1250) — compile-verified
//
#include <hip/hip_runtime.h>
#include <hip/hip_bf16.h>

// ---------------------------------------------------------------------------
// SST input layer for gfx1250 (MI455X).  Pipeline of small kernels, all on
// `stream`, all deterministic.  Data-dependent sizes live in a device-side
// control block (ctrl[]):
//   ctrl[0]=n1 (after drop0)  ctrl[1]=n2 (final)  ctrl[2]=numWin0
//   ctrl[3]=numWin1           ctrl[8]=N
// Output layout (floats): feats[n2*128] | posw0[nw0*120*128] |
//   posw1[nw1*120*128] | mask0[nw0*120] | mask1[nw1*120] | keep[n2]
// ---------------------------------------------------------------------------

#define WMAXW 8192   // upper bound on #windows (actual: 2*40*40*2 = 6400)
#define CH    512    // chunk length per ranking block
#define MT    120    // max tokens per window
#define CDIM  128    // feature dim

typedef __attribute__((ext_vector_type(2))) float v2f;
typedef __attribute__((ext_vector_type(8))) float v8f;

// ---------------- utility --------------------------------------------------
__global__ void k_zero_i32(int* __restrict__ p, long n) {
  long g = (long)blockIdx.x * blockDim.x + threadIdx.x;
  if (g < n) p[g] = 0;
}
__global__ void k_zero_f32(float* __restrict__ p, long n) {
  long g = (long)blockIdx.x * blockDim.x + threadIdx.x;
  if (g < n) p[g] = 0.0f;
}
__global__ void k_init_ctrl(int* ctrl, int N) {
  if (threadIdx.x == 0 && blockIdx.x == 0) ctrl[8] = N;
}

// ---------------- window ids ----------------------------------------------
__global__ void k_compute_b(const int* __restrict__ coors,
                            const int* __restrict__ gs, int N,
                            int* __restrict__ b0, int* __restrict__ b1) {
  int i = blockIdx.x * blockDim.x + threadIdx.x;
  if (i >= N) return;
  int gx = gs[0], gy = gs[1], gz = gs[2];
  int mx = (gx + 11) / 12 + 1;
  int my = (gy + 11) / 12 + 1;
  int mz = gz + 1;                      // wz == 1
  int per = mx * my * mz;
  int bt = coors[i * 4 + 0];
  int z  = coors[i * 4 + 1];
  int y  = coors[i * 4 + 2];
  int x  = coors[i * 4 + 3];
  b0[i] = bt * per + (x / 12) * (my * mz) + (y / 12) * mz + z;
  int xs = x + 6, ys = y + 6;           // shift = (6,6,0)
  b1[i] = bt * per + (xs / 12) * (my * mz) + (ys / 12) * mz + z;
}

// ---------------- stable within-window rank --------------------------------
// phase A: per-chunk histogram (atomic sums -> deterministic)
__global__ void k_hist(const int* __restrict__ bwin, const int* __restrict__ Lp,
                       int* __restrict__ bh) {
  int g = blockIdx.x * blockDim.x + threadIdx.x;
  if (g >= *Lp) return;
  int w = bwin[g];
  if ((unsigned)w < WMAXW)
    atomicAdd(&bh[(size_t)(g / CH) * WMAXW + w], 1);
}
// phase B: exclusive scan over chunks per window (in place) + totals
__global__ void k_scan_blocks(int* __restrict__ bh, int* __restrict__ totals,
                              int NB) {
  int w = blockIdx.x * blockDim.x + threadIdx.x;
  if (w >= WMAXW) return;
  int run = 0;
  for (int b = 0; b < NB; ++b) {
    size_t ix = (size_t)b * WMAXW + w;
    int t = bh[ix];
    bh[ix] = run;
    run += t;
  }
  totals[w] = run;
}
// phase C: one wave (32 threads) per chunk, in-order; LDS counters + shfl
// leader election give the deterministic order-of-appearance rank.
__global__ void k_rank(const int* __restrict__ bwin, const int* __restrict__ Lp,
                       const int* __restrict__ bh, int* __restrict__ rank) {
  __shared__ int cnt[WMAXW];            // 32 KB
  int L = *Lp;
  int base = blockIdx.x * CH;
  if (base >= L) return;
  int lane = threadIdx.x;               // block == 1 wave of 32
  for (int j = lane; j < WMAXW; j += 32) cnt[j] = 0;
  __syncthreads();
  const int* off = bh + (size_t)blockIdx.x * WMAXW;
  for (int s = 0; s < CH / 32; ++s) {
    int i = base + s * 32 + lane;
    bool act = i < L;
    int w = act ? bwin[i] : -1;
    int before = 0, total = 0, leader = -1;
    for (int k = 0; k < 32; ++k) {
      int wk = __shfl(w, k);
      if (wk == w) {
        total++;
        if (k < lane) before++;
        if (leader < 0) leader = k;
      }
    }
    int old = 0;
    if (act && before == 0 && (unsigned)w < WMAXW) {
      old = cnt[w];
      cnt[w] = old + total;             // distinct leaders -> distinct w
    }
    old = __shfl(old, leader);
    if (act && (unsigned)w < WMAXW) rank[i] = off[w] + old + before;
    __syncthreads();
  }
}

// ---------------- stable compaction ---------------------------------------
__device__ __forceinline__ bool keep_pred(int g, const int* rank,
                                          const int* bwin, const int* totals) {
  int w = bwin[g];
  int npv = ((unsigned)w < WMAXW) ? totals[w] : 0;
  return (rank[g] < MT) && (npv < 100000);
}
__global__ void k_cpt_count(const int* __restrict__ rank,
                            const int* __restrict__ bwin,
                            const int* __restrict__ totals,
                            const int* __restrict__ Lp,
                            int* __restrict__ bcnt) {
  int L = *Lp;
  int g = blockIdx.x * 256 + threadIdx.x;
  bool p = (g < L) && keep_pred(g, rank, bwin, totals);
  unsigned long long bal = __ballot(p);
  if ((threadIdx.x & 31) == 0) atomicAdd(&bcnt[blockIdx.x], __popcll(bal));
}
__global__ void k_cpt_scan(int* __restrict__ bcnt, int* __restrict__ boff,
                           int NB2, int* __restrict__ outTotal) {
  if (threadIdx.x == 0 && blockIdx.x == 0) {
    int s = 0;
    for (int i = 0; i < NB2; ++i) { boff[i] = s; s += bcnt[i]; }
    *outTotal = s;
  }
}
__global__ void k_cpt_write(const int* __restrict__ rank,
                            const int* __restrict__ bwin,
                            const int* __restrict__ totals,
                            const int* __restrict__ Lp,
                            const int* __restrict__ boff,
                            const int* __restrict__ vals,  // may be nullptr
                            int* __restrict__ outIdx) {
  int L = *Lp;
  int g = blockIdx.x * 256 + threadIdx.x;
  bool p = (g < L) && keep_pred(g, rank, bwin, totals);
  unsigned long long bal = __ballot(p);
  int lane = threadIdx.x & 31, wid = threadIdx.x >> 5;
  int within = __popcll(bal & ((1ull << lane) - 1ull));
  __shared__ int woff[8];
  if (lane == 0) woff[wid] = __popcll(bal);
  __syncthreads();
  if (threadIdx.x == 0) {
    int s = 0;
    for (int k = 0; k < 8; ++k) { int t = woff[k]; woff[k] = s; s += t; }
  }
  __syncthreads();
  if (p) {
    int pos = boff[blockIdx.x] + woff[wid] + within;
    outIdx[pos] = vals ? vals[g] : g;
  }
}

// ---------------- gather of b / misc ---------------------------------------
__global__ void k_gather_i32(const int* __restrict__ idx,
                             const int* __restrict__ src,
                             const int* __restrict__ Lp, int* __restrict__ dst) {
  int g = blockIdx.x * blockDim.x + threadIdx.x;
  if (g >= *Lp) return;
  dst[g] = src[idx[g]];
}
__global__ void k_winflag(const int* __restrict__ fb, const int* __restrict__ Lp,
                          int* __restrict__ flag) {
  int g = blockIdx.x * blockDim.x + threadIdx.x;
  if (g >= *Lp) return;
  int w = fb[g];
  if ((unsigned)w < WMAXW) flag[w] = 1;   // idempotent -> deterministic
}
__global__ void k_winscan(const int* __restrict__ f0, const int* __restrict__ f1,
                          int* __restrict__ id0, int* __restrict__ id1,
                          int* __restrict__ ctrl) {
  if (threadIdx.x || blockIdx.x) return;
  int s = 0;
  for (int w = 0; w < WMAXW; ++w) { id0[w] = s; s += f0[w]; }
  ctrl[2] = s;
  s = 0;
  for (int w = 0; w < WMAXW; ++w) { id1[w] = s; s += f1[w]; }
  ctrl[3] = s;
}

// ---------------- positional-embedding table via WMMA ----------------------
// table[p][2k]   = sin((p-6) * 10000^(-k/32))
// table[p][2k+1] = cos((p-6) * 10000^(-k/32)),  p in [0,12), k in [0,32)
// Outer product (p-6) x 10000^(-k/32) done with V_WMMA_F32_16X16X4_F32:
// A is 16x4 with only K=0 populated, B is 4x16 with only row K=0 populated.
__global__ void k_table(float* __restrict__ table) {
  int lane = threadIdx.x;               // exactly one wave of 32
  v2f a;
  a[0] = (lane < 16) ? (float)lane - 6.0f : 0.0f;  // A[M=lane][K=0]
  a[1] = 0.0f;                                     // K=1 / K=3 = 0
  v8f c = {};
  for (int h = 0; h < 2; ++h) {
    int kf = (lane & 15) + 16 * h;
    v2f b;
    b[0] = (lane < 16) ? powf(10000.0f, -(float)kf / 32.0f) : 0.0f; // B[K=0][N]
    b[1] = 0.0f;
    v8f d = __builtin_amdgcn_wmma_f32_16x16x4_f32(
        false, a, false, b, (short)0, c, false, false);
    int mbase = (lane >> 4) * 8;
    int n = lane & 15;
    int k = n + 16 * h;
    for (int r = 0; r < 8; ++r) {
      int m = mbase + r;
      if (m < 12) {
        float ph = d[r];                // (m-6) * 10000^(-k/32)
        table[m * 64 + 2 * k]     = sinf(ph);
        table[m * 64 + 2 * k + 1] = cosf(ph);
      }
    }
  }
}

// ---------------- output assembly ------------------------------------------
__global__ void k_mask_ones(float* __restrict__ out, const int* __restrict__ ctrl) {
  long g = (long)blockIdx.x * blockDim.x + threadIdx.x;
  int n2 = ctrl[1], nw0 = ctrl[2], nw1 = ctrl[3];
  long nmask = (long)(nw0 + nw1) * MT;
  if (g >= nmask) return;
  long base = (long)n2 * CDIM + (long)nw0 * MT * CDIM + (long)nw1 * MT * CDIM;
  out[base + g] = 1.0f;                 // key_mask default: True (empty slot)
}

__global__ void k_scatter(const float* __restrict__ feats,
                          const int* __restrict__ coors,
                          const int* __restrict__ ctrl,
                          const int* __restrict__ idxB,
                          const int* __restrict__ fb0,
                          const int* __restrict__ fb1,
                          const int* __restrict__ inner0,
                          const int* __restrict__ inner1,
                          const int* __restrict__ winId0,
                          const int* __restrict__ winId1,
                          const float* __restrict__ table,
                          float* __restrict__ out) {
  long g = (long)blockIdx.x * blockDim.x + threadIdx.x;
  int t = (int)(g >> 5);
  int lane = (int)(g & 31);
  int n2 = ctrl[1], nw0 = ctrl[2], nw1 = ctrl[3];
  if (t >= n2) return;
  long off_pw0  = (long)n2 * CDIM;
  long off_pw1  = off_pw0 + (long)nw0 * MT * CDIM;
  long off_km0  = off_pw1 + (long)nw1 * MT * CDIM;
  long off_km1  = off_km0 + (long)nw0 * MT;
  long off_keep = off_km1 + (long)nw1 * MT;

  int src = idxB[t];
  int d = lane * 4;

  // prefetch a future token's feature row (global_prefetch_b8)
  int tn = t + 64;
  if (tn < n2) __builtin_prefetch(feats + (size_t)idxB[tn] * CDIM + d, 0, 0);

  // features gather (coalesced 128-bit)
  float4 f = *(const float4*)(feats + (size_t)src * CDIM + d);
  *(float4*)(out + (long)t * CDIM + d) = f;

  int x = coors[src * 4 + 3];
  int y = coors[src * 4 + 2];
  int px0 = x % 12,       py0 = y % 12;
  int px1 = (x + 6) % 12, py1 = (y + 6) % 12;

  long slot0 = (long)winId0[fb0[t]] * MT + inner0[t];
  long slot1 = (long)winId1[fb1[t]] * MT + inner1[t];

  const float* t0 = table + ((d < 64) ? px0 : py0) * 64 + (d & 63);
  *(float4*)(out + off_pw0 + slot0 * CDIM + d) = *(const float4*)t0;
  const float* t1 = table + ((d < 64) ? px1 : py1) * 64 + (d & 63);
  *(float4*)(out + off_pw1 + slot1 * CDIM + d) = *(const float4*)t1;

  if (lane == 0) {
    out[off_km0 + slot0] = 0.0f;        // token present -> not masked
    out[off_km1 + slot1] = 0.0f;
    out[off_keep + t] = (float)src;
  }
}

// ---------------------------------------------------------------------------
extern "C" void kernel_launch(void* const* d_in, const int* in_sizes, int n_in,
                              void* d_out, int out_size, void* d_ws, size_t ws_size,
                              hipStream_t stream) {
  const float* feats = (const float*)d_in[0];
  const int*   coors = (const int*)d_in[1];
  const int*   grid  = (const int*)d_in[3];
  int N = in_sizes[2];                   // voxel_shuffle_inds length
  float* out = (float*)d_out;

  int NBmax = (N + CH - 1) / CH;
  int NB2   = (N + 255) / 256;

  // workspace carve-up (all int32-sized slots, 16B-aligned regions)
  int* W = (int*)d_ws;
  size_t o = 0;
  auto alloc = [&](size_t n) { int* p = W + o; o += (n + 3) & ~(size_t)3; return p; };
  int* b0      = alloc(N);
  int* b1      = alloc(N);
  int* rankA   = alloc(N);
  int* listB   = alloc(N);
  int* idxA    = alloc(N);
  int* idxB    = alloc(N);
  int* inner0  = alloc(N);
  int* inner1  = alloc(N);
  int* fb0     = alloc(N);
  int* fb1     = alloc(N);
  int* totalsA = alloc(WMAXW);
  int* totalsB = alloc(WMAXW);
  int* winFlag0 = alloc(WMAXW);
  int* winFlag1 = alloc(WMAXW);
  int* winId0  = alloc(WMAXW);
  int* winId1  = alloc(WMAXW);
  int* blockCnt = alloc(NB2);
  int* blockOff = alloc(NB2);
  int* ctrl    = alloc(32);
  float* table = (float*)alloc(12 * 64);
  int* bh      = alloc((size_t)NBmax * WMAXW);
  (void)ws_size;

  const int T = 256;
  int gN   = (N + T - 1) / T;
  int gWin = (WMAXW + T - 1) / T;
  long bhCnt = (long)NBmax * WMAXW;
  int gBH  = (int)((bhCnt + T - 1) / T);
  int gCpt = NB2;

  k_init_ctrl<<<1, 32, 0, stream>>>(ctrl, N);
  k_table<<<1, 32, 0, stream>>>(table);
  k_compute_b<<<gN, T, 0, stream>>>(coors, grid, N, b0, b1);

  // ---- pass 1: rank over b0 (L = N at ctrl[8]) ----
  k_zero_i32<<<gBH, T, 0, stream>>>(bh, bhCnt);
  k_hist<<<gN, T, 0, stream>>>(b0, ctrl + 8, bh);
  k_scan_blocks<<<gWin, T, 0, stream>>>(bh, totalsA, NBmax);
  k_rank<<<NBmax, 32, 0, stream>>>(b0, ctrl + 8, bh, rankA);
  // ---- compaction 1 -> idxA, n1 = ctrl[0] ----
  k_zero_i32<<<(2 * NB2 + T - 1) / T, T, 0, stream>>>(blockCnt, 2 * NB2);
  k_cpt_count<<<gCpt, T, 0, stream>>>(rankA, b0, totalsA, ctrl + 8, blockCnt);
  k_cpt_scan<<<1, 32, 0, stream>>>(blockCnt, blockOff, NB2, ctrl + 0);
  k_cpt_write<<<gCpt, T, 0, stream>>>(rankA, b0, totalsA, ctrl + 8, blockOff,
                                      nullptr, idxA);
  // listB = b1[idxA]
  k_gather_i32<<<gN, T, 0, stream>>>(idxA, b1, ctrl + 0, listB);

  // ---- pass 2: rank over listB (L = n1) ----
  k_zero_i32<<<gBH, T, 0, stream>>>(bh, bhCnt);
  k_hist<<<gN, T, 0, stream>>>(listB, ctrl + 0, bh);
  k_scan_blocks<<<gWin, T, 0, stream>>>(bh, totalsB, NBmax);
  k_rank<<<NBmax, 32, 0, stream>>>(listB, ctrl + 0, bh, rankA);
  // ---- compaction 2 -> idxB, n2 = ctrl[1] ----
  k_zero_i32<<<(2 * NB2 + T - 1) / T, T, 0, stream>>>(blockCnt, 2 * NB2);
  k_cpt_count<<<gCpt, T, 0, stream>>>(rankA, listB, totalsB, ctrl + 0, blockCnt);
  k_cpt_scan<<<1, 32, 0, stream>>>(blockCnt, blockOff, NB2, ctrl + 1);
  k_cpt_write<<<gCpt, T, 0, stream>>>(rankA, listB, totalsB, ctrl + 0, blockOff,
                                      idxA, idxB);

  // final per-token window ids
  k_gather_i32<<<gN, T, 0, stream>>>(idxB, b0, ctrl + 1, fb0);
  k_gather_i32<<<gN, T, 0, stream>>>(idxB, b1, ctrl + 1, fb1);

  // dense window numbering (sorted-unique order == ascending window id)
  k_zero_i32<<<(2 * WMAXW + T - 1) / T, T, 0, stream>>>(winFlag0, 2 * WMAXW);
  k_winflag<<<gN, T, 0, stream>>>(fb0, ctrl + 1, winFlag0);
  k_winflag<<<gN, T, 0, stream>>>(fb1, ctrl + 1, winFlag1);
  k_winscan<<<1, 32, 0, stream>>>(winFlag0, winFlag1, winId0, winId1, ctrl);

  // ---- pass 3: inner rank over fb0 (L = n2) ----
  k_zero_i32<<<gBH, T, 0, stream>>>(bh, bhCnt);
  k_hist<<<gN, T, 0, stream>>>(fb0, ctrl + 1, bh);
  k_scan_blocks<<<gWin, T, 0, stream>>>(bh, totalsA, NBmax);
  k_rank<<<NBmax, 32, 0, stream>>>(fb0, ctrl + 1, bh, inner0);
  // ---- pass 4: inner rank over fb1 (L = n2) ----
  k_zero_i32<<<gBH, T, 0, stream>>>(bh, bhCnt);
  k_hist<<<gN, T, 0, stream>>>(fb1, ctrl + 1, bh);
  k_scan_blocks<<<gWin, T, 0, stream>>>(bh, totalsA, NBmax);
  k_rank<<<NBmax, 32, 0, stream>>>(fb1, ctrl + 1, bh, inner1);

  // ---- outputs ----
  k_zero_f32<<<(int)(((long)out_size + T - 1) / T), T, 0, stream>>>(out, out_size);
  long maskMax = 2L * WMAXW * MT;
  k_mask_ones<<<(int)((maskMax + T - 1) / T), T, 0, stream>>>(out, ctrl);
  long scatMax = (long)N * 32;
  k_scatter<<<(int)((scatMax + T - 1) / T), T, 0, stream>>>(
      feats, coors, ctrl, idxB, fb0, fb1, inner0, inner1, winId0, winId1,
      table, out);
}